// TverskyMultiHeadAttentionDropIn_90666759618852
// MI455X (gfx1250) — compile-verified
//
#include <hip/hip_runtime.h>
#include <hip/hip_bf16.h>

// ---------------------------------------------------------------------------
// TverskyMultiHeadAttention for MI455X (gfx1250), fp32 throughout.
// D=512, H=8 (dk=64), K=128 features, B=2, S=512  ->  N = B*S = 1024 tokens.
//
// GEMM-shaped math: V_WMMA_F32_16X16X4_F32 (fp32 WMMA).
// LDS staging: GLOBAL_LOAD_ASYNC_TO_LDS_B128 (ASYNCcnt) — CDNA5 direct
// global->LDS path, no VGPR round trip; s_wait_asynccnt + barrier to publish.
// Tversky difference terms are irreducibly elementwise over (N x O x K) and
// run as a float4-vectorized LDS VALU loop fused with the WMMA intersection.
// ---------------------------------------------------------------------------

typedef __attribute__((ext_vector_type(2))) float v2f;
typedef __attribute__((ext_vector_type(8))) float v8f;

#define DMODEL 512
#define KFEAT  128
#define SEQ    512
#define NBATCH 2
#define NHEAD  8
#define DK     64
#define NTOK   (NBATCH * SEQ)   // 1024

__device__ __forceinline__ v8f wmma_f32(v2f a, v2f b, v8f c) {
  // D = A(16x4) * B(4x16) + C, fp32.
  // (neg_a, A, neg_b, B, c_mod, C, reuse_a, reuse_b)
  return __builtin_amdgcn_wmma_f32_16x16x4_f32(false, a, false, b, (short)0, c,
                                               false, false);
}

// Async global -> LDS copy of 16 bytes (per lane). LDS operand is the
// wave-relative LDS byte offset (low 32 bits of the generic shared address).
__device__ __forceinline__ void async_copy_b128(const void* gaddr,
                                                void* lds_ptr) {
  const unsigned lds = (unsigned)(uintptr_t)lds_ptr;
  asm volatile("global_load_async_to_lds_b128 %0, %1, off"
               :
               : "v"(lds), "v"(gaddr)
               : "memory");
}

__device__ __forceinline__ void async_wait0() {
  asm volatile("s_wait_asynccnt 0x0" ::: "memory");
}

// ---------------------------------------------------------------------------
// Kernel 1: F(M x 128) = A(M x 512) @ W(512 x 128), fp32 WMMA.
// Block = 256 threads (8 waves); waves share one 16-column W panel staged in
// LDS by async b128 copies. Wave w owns m-tile (blockIdx.x*128 + w*16).
// grid = (M/128, 128/16).
// ---------------------------------------------------------------------------
__global__ __launch_bounds__(256) void feat_gemm(const float* __restrict__ A,
                                                 const float* __restrict__ W,
                                                 float* __restrict__ F) {
  __shared__ float sw[DMODEL][16];   // 32 KB: W[:, n0..n0+15]

  const int t  = threadIdx.x;
  const int m0 = blockIdx.x * 128;
  const int n0 = blockIdx.y * 16;

  // stage W panel: 512 rows x 4 float4s = 2048 copies / 256 threads = 8 each
  for (int i = t; i < DMODEL * 4; i += 256) {
    const int r = i >> 2, c = (i & 3) * 4;
    async_copy_b128(W + (size_t)r * KFEAT + n0 + c, &sw[r][c]);
  }
  async_wait0();
  __syncthreads();

  const int w = t >> 5, lane = t & 31;
  const int half = lane >> 4, l15 = lane & 15, koff = half * 2;
  const int mt = m0 + w * 16;

  const float* arow = A + (size_t)(mt + l15) * DMODEL;
  v8f acc = {};
  for (int k = 0; k < DMODEL; k += 4) {
    v2f a, b;
    const float2 a2 = *(const float2*)(arow + k + koff);
    a.x = a2.x;
    a.y = a2.y;
    b.x = sw[k + koff][l15];
    b.y = sw[k + koff + 1][l15];
    acc = wmma_f32(a, b, acc);
  }
  const int mb = mt + half * 8;
#pragma unroll
  for (int v = 0; v < 8; ++v) {
    F[(size_t)(mb + v) * KFEAT + n0 + l15] = acc[v];
  }
}

// ---------------------------------------------------------------------------
// Kernel 2: Tversky contrast projection.
// OUT[n,o] = theta * sum_k relu(xf)relu(pf)
//          - alpha * sum_k relu(xf-pf)*[xf>0]
//          - beta  * sum_k relu(pf-xf)*[pf>0]
// Macro-tile 32(n) x 64(o), staged in LDS via async b128; 8 waves, each a
// 16x16 subtile: WMMA intersection + float4 VALU difference loop.
// grid = (Nrows/32, 512/64), block = 256.
// ---------------------------------------------------------------------------
__device__ __forceinline__ void tv_acc(float a, float b, float& dab,
                                       float& dba) {
  const float d = a - b;
  dab += (a > 0.f) ? fmaxf(d, 0.f) : 0.f;
  dba += (b > 0.f) ? fmaxf(-d, 0.f) : 0.f;
}

__global__ __launch_bounds__(256) void tversky_kernel(
    const float* __restrict__ XF, const float* __restrict__ PFm,
    const float* __restrict__ tp, int pidx, float* __restrict__ OUT) {
  // stride 132: float4-aligned rows (528 B) and conflict-free column access
  __shared__ float sx[32][132];
  __shared__ float sp[64][132];

  const int n0 = blockIdx.x * 32;
  const int o0 = blockIdx.y * 64;
  const int t  = threadIdx.x;

  for (int i = t; i < 32 * 32; i += 256) {
    const int r = i >> 5, c = (i & 31) * 4;
    async_copy_b128(XF + (size_t)(n0 + r) * KFEAT + c, &sx[r][c]);
  }
  for (int i = t; i < 64 * 32; i += 256) {
    const int r = i >> 5, c = (i & 31) * 4;
    async_copy_b128(PFm + (size_t)(o0 + r) * KFEAT + c, &sp[r][c]);
  }
  async_wait0();
  __syncthreads();

  const int w = t >> 5, lane = t & 31;
  const int half = lane >> 4, l15 = lane & 15, koff = half * 2;
  const int wr = w >> 2;   // 0..1 -> n subtile
  const int wc = w & 3;    // 0..3 -> o subtile

  // --- intersection: relu(xf) @ relu(pf)^T via WMMA ---
  v8f acc = {};
  for (int k = 0; k < KFEAT; k += 4) {
    v2f a, b;
    a.x = fmaxf(sx[wr * 16 + l15][k + koff], 0.f);
    a.y = fmaxf(sx[wr * 16 + l15][k + koff + 1], 0.f);
    b.x = fmaxf(sp[wc * 16 + l15][k + koff], 0.f);
    b.y = fmaxf(sp[wc * 16 + l15][k + koff + 1], 0.f);
    acc = wmma_f32(a, b, acc);
  }

  // --- difference reductions (elementwise over K), same C/D slot mapping ---
  const int ol = wc * 16 + l15;        // o column this lane owns
  const int mb = wr * 16 + half * 8;   // first of 8 n rows this lane owns
  float dab[8] = {0.f, 0.f, 0.f, 0.f, 0.f, 0.f, 0.f, 0.f};
  float dba[8] = {0.f, 0.f, 0.f, 0.f, 0.f, 0.f, 0.f, 0.f};
  for (int k = 0; k < KFEAT; k += 4) {
    const float4 bq = *(const float4*)&sp[ol][k];
#pragma unroll
    for (int v = 0; v < 8; ++v) {
      const float4 aq = *(const float4*)&sx[mb + v][k];
      tv_acc(aq.x, bq.x, dab[v], dba[v]);
      tv_acc(aq.y, bq.y, dab[v], dba[v]);
      tv_acc(aq.z, bq.z, dab[v], dba[v]);
      tv_acc(aq.w, bq.w, dab[v], dba[v]);
    }
  }

  const float theta = tp[pidx * 3 + 0];
  const float alpha = tp[pidx * 3 + 1];
  const float beta  = tp[pidx * 3 + 2];
#pragma unroll
  for (int v = 0; v < 8; ++v) {
    OUT[(size_t)(n0 + mb + v) * DMODEL + (o0 + ol)] =
        theta * acc[v] - alpha * dab[v] - beta * dba[v];
  }
}

// ---------------------------------------------------------------------------
// Kernel 3: attention for one (b,h) and 16 query rows.
// Phase 1: scores = Q K^T / sqrt(dk) (WMMA), mask fused, stored in LDS.
// Phase 2: softmax over 512 keys (16 threads per row).
// Phase 3: ctx = P @ V (WMMA, A-fragments from LDS scores).
// grid = (S/16, B*H), block = 256.
// ---------------------------------------------------------------------------
__global__ __launch_bounds__(256) void attn_kernel(
    const float* __restrict__ QP, const float* __restrict__ KP,
    const float* __restrict__ VP, const int* __restrict__ maskp,
    float* __restrict__ CTX) {
  __shared__ float s_sc[16][516];   // score rows, 16B-aligned stride, conflict-free
  __shared__ float s_red[16][16];
  __shared__ float s_row[16];

  const int bh = blockIdx.y;
  const int b = bh >> 3, h = bh & 7;
  const int q0 = blockIdx.x * 16;

  const int t = threadIdx.x;
  const int w = t >> 5, lane = t & 31;
  const int half = lane >> 4, l15 = lane & 15, koff = half * 2;

  const float* Qb = QP + (size_t)(b * SEQ) * DMODEL + h * DK;
  const float* Kb = KP + (size_t)(b * SEQ) * DMODEL + h * DK;
  const float* Vb = VP + (size_t)(b * SEQ) * DMODEL + h * DK;
  const float neg_inf = -__builtin_inff();

  // ---- phase 1: scores (each wave handles 4 of the 32 key tiles) ----
  for (int ct = w; ct < SEQ / 16; ct += 8) {
    v8f acc = {};
    const int qrow = q0 + l15;       // A-fragment row
    const int key  = ct * 16 + l15;  // B-fragment column
    for (int k = 0; k < DK; k += 4) {
      v2f a, bb;
      const float2 qa = *(const float2*)(Qb + (size_t)qrow * DMODEL + k + koff);
      a.x = qa.x;
      a.y = qa.y;
      const float2 kb = *(const float2*)(Kb + (size_t)key * DMODEL + k + koff);
      bb.x = kb.x;
      bb.y = kb.y;
      acc = wmma_f32(a, bb, acc);
    }
#pragma unroll
    for (int v = 0; v < 8; ++v) {
      const int qr = half * 8 + v;       // local query row
      const int kg = ct * 16 + l15;      // key index
      const int m = maskp[((size_t)b * SEQ + (q0 + qr)) * SEQ + kg];
      s_sc[qr][kg] = m ? acc[v] * 0.125f : neg_inf;  // 1/sqrt(64)
    }
  }
  __syncthreads();

  // ---- phase 2: softmax, 16 threads per row, 32 keys per thread ----
  {
    const int r = t >> 4;
    const int c = t & 15;
    float mx = neg_inf;
    for (int j = c * 32; j < c * 32 + 32; ++j) mx = fmaxf(mx, s_sc[r][j]);
    s_red[r][c] = mx;
    __syncthreads();
    if (c == 0) {
      float m2 = neg_inf;
      for (int j = 0; j < 16; ++j) m2 = fmaxf(m2, s_red[r][j]);
      s_row[r] = m2;
    }
    __syncthreads();
    const float rm = s_row[r];
    float sum = 0.f;
    for (int j = c * 32; j < c * 32 + 32; ++j) {
      const float e = __expf(s_sc[r][j] - rm);
      s_sc[r][j] = e;
      sum += e;
    }
    s_red[r][c] = sum;
    __syncthreads();
    if (c == 0) {
      float s2 = 0.f;
      for (int j = 0; j < 16; ++j) s2 += s_red[r][j];
      s_row[r] = 1.f / s2;
    }
    __syncthreads();
    const float inv = s_row[r];
    for (int j = c * 32; j < c * 32 + 32; ++j) s_sc[r][j] *= inv;
  }
  __syncthreads();

  // ---- phase 3: ctx = P(16x512) @ V(512x64); waves 0..3, one 16-col tile each
  if (w < 4) {
    v8f acc = {};
    for (int k = 0; k < SEQ; k += 4) {
      v2f a, bb;
      a.x = s_sc[l15][k + koff];
      a.y = s_sc[l15][k + koff + 1];
      const int kr = k + koff;
      bb.x = Vb[(size_t)kr * DMODEL + w * 16 + l15];
      bb.y = Vb[(size_t)(kr + 1) * DMODEL + w * 16 + l15];
      acc = wmma_f32(a, bb, acc);
    }
#pragma unroll
    for (int v = 0; v < 8; ++v) {
      const int qr = half * 8 + v;
      CTX[((size_t)(b * SEQ + q0 + qr)) * DMODEL + h * DK + w * 16 + l15] =
          acc[v];
    }
  }
}

// ---------------------------------------------------------------------------
// Host orchestration.
// ---------------------------------------------------------------------------
extern "C" void kernel_launch(void* const* d_in, const int* in_sizes, int n_in,
                              void* d_out, int out_size, void* d_ws,
                              size_t ws_size, hipStream_t stream) {
  const float* x     = (const float*)d_in[0];  // (B,S,D)
  const int*   mask  = (const int*)d_in[1];    // (B,1,S,S)
  const float* omega = (const float*)d_in[2];  // (D,K)
  const float* Pq    = (const float*)d_in[3];  // (D,D)
  const float* Pk    = (const float*)d_in[4];
  const float* Pv    = (const float*)d_in[5];
  const float* Po    = (const float*)d_in[6];
  const float* tp    = (const float*)d_in[7];  // (4,3)

  float* ws  = (float*)d_ws;
  float* XF  = ws;                          // 1024 x 128
  float* PF  = XF + (size_t)NTOK * KFEAT;   // 4 x (512 x 128)
  float* QP  = PF + (size_t)4 * DMODEL * KFEAT;  // 1024 x 512
  float* KPp = QP + (size_t)NTOK * DMODEL;
  float* VPp = KPp + (size_t)NTOK * DMODEL;
  float* CTX = VPp + (size_t)NTOK * DMODEL;
  float* CF  = CTX + (size_t)NTOK * DMODEL;  // 1024 x 128
  // total: ~10.5 MB of fp32 workspace (fully L2-resident on MI455X)

  const dim3 blk256(256);
  const size_t pfs = (size_t)DMODEL * KFEAT;  // 512*128 per prototype bank

  // 1) feature memberships
  feat_gemm<<<dim3(NTOK / 128, KFEAT / 16), blk256, 0, stream>>>(x, omega, XF);
  feat_gemm<<<dim3(DMODEL / 128, KFEAT / 16), blk256, 0, stream>>>(Pq, omega, PF + 0 * pfs);
  feat_gemm<<<dim3(DMODEL / 128, KFEAT / 16), blk256, 0, stream>>>(Pk, omega, PF + 1 * pfs);
  feat_gemm<<<dim3(DMODEL / 128, KFEAT / 16), blk256, 0, stream>>>(Pv, omega, PF + 2 * pfs);
  feat_gemm<<<dim3(DMODEL / 128, KFEAT / 16), blk256, 0, stream>>>(Po, omega, PF + 3 * pfs);

  // 2) Tversky projections -> Q, K, V (stored (N, h*dk+d) row-major)
  tversky_kernel<<<dim3(NTOK / 32, DMODEL / 64), blk256, 0, stream>>>(XF, PF + 0 * pfs, tp, 0, QP);
  tversky_kernel<<<dim3(NTOK / 32, DMODEL / 64), blk256, 0, stream>>>(XF, PF + 1 * pfs, tp, 1, KPp);
  tversky_kernel<<<dim3(NTOK / 32, DMODEL / 64), blk256, 0, stream>>>(XF, PF + 2 * pfs, tp, 2, VPp);

  // 3) attention per (b,h)
  attn_kernel<<<dim3(SEQ / 16, NBATCH * NHEAD), blk256, 0, stream>>>(QP, KPp, VPp, mask, CTX);

  // 4) output projection
  feat_gemm<<<dim3(NTOK / 128, KFEAT / 16), blk256, 0, stream>>>(CTX, omega, CF);
  tversky_kernel<<<dim3(NTOK / 32, DMODEL / 64), blk256, 0, stream>>>(CF, PF + 3 * pfs, tp, 3, (float*)d_out);
}